// RouteNetLiteLayer_52664888984238
// MI455X (gfx1250) — compile-verified
//
#include <hip/hip_runtime.h>
#include <hip/hip_bf16.h>
#include <climits>

// ---------------------------------------------------------------------------
// RouteNetLite GAT layer for MI455X (gfx1250, wave32, WMMA)
//   q,k,v = h @ W{q,k,v}^T         (WMMA f32 16x16x4)
//   scores_e = <q[dst], k[src]> / sqrt(128)
//   segment softmax over dst       (ordered-int atomic max + f32 atomic add)
//   agg = segsum(alpha * v[src])   (f32 atomics)
//   out = relu(agg @ Wp^T + bp + h) (WMMA f32 16x16x4, fused epilogue)
// ---------------------------------------------------------------------------

#define DIM 128

typedef float v2f __attribute__((ext_vector_type(2)));
typedef float v8f __attribute__((ext_vector_type(8)));

// ---------------------------------------------------------------------------
// init: zero agg[N*128]; maxkey[N] = INT_MIN; sum[N] = 0
// ---------------------------------------------------------------------------
__global__ void rnl_init_kernel(float* __restrict__ agg,
                                int*   __restrict__ maxkey,
                                float* __restrict__ sumv,
                                int N) {
    int i = blockIdx.x * blockDim.x + threadIdx.x;
    size_t total = (size_t)N * DIM;
    if ((size_t)i < total) agg[i] = 0.0f;
    if (i < N) {
        maxkey[i] = INT_MIN;   // identity for ordered-int float max
        sumv[i]   = 0.0f;
    }
}

// ---------------------------------------------------------------------------
// WMMA core: one wave computes a 16x16 tile of  X @ W^T  (X: [N,128], W: [128,128]).
// A-matrix layout (16x4 f32): lanes 0-15 hold row M=lane, VGPR0=K0,VGPR1=K1;
// lanes 16-31 hold row M=lane-16, VGPR0=K2,VGPR1=K3  -> one b64 load per lane.
// B = W^T, i.e. B[k][n] = W[col0+n][k]; same striping over K -> one b64 load.
// C/D layout: VGPR r: lanes 0-15 -> (row0+r, col0+lane), lanes 16-31 -> (row0+r+8).
// ---------------------------------------------------------------------------
__device__ __forceinline__ v8f wmma_tile_xwT(const float* __restrict__ X,
                                             const float* __restrict__ W,
                                             int row0, int col0,
                                             int lane) {
    const int half = lane >> 4;    // 0: K pair (0,1)   1: K pair (2,3)
    const int l16  = lane & 15;

    const float* aptr = X + (size_t)(row0 + l16) * DIM + 2 * half;
    const float* bptr = W + (size_t)(col0 + l16) * DIM + 2 * half;

    v8f c = {0.f, 0.f, 0.f, 0.f, 0.f, 0.f, 0.f, 0.f};
#pragma unroll
    for (int kk = 0; kk < DIM; kk += 4) {
        v2f a = *(const v2f*)(aptr + kk);   // global_load_b64
        v2f b = *(const v2f*)(bptr + kk);   // global_load_b64
        // v_wmma_f32_16x16x4_f32  (8-arg form: neg_a, A, neg_b, B, c_mod, C, reuse_a, reuse_b)
        c = __builtin_amdgcn_wmma_f32_16x16x4_f32(false, a, false, b,
                                                  (short)0, c, false, false);
    }
    return c;
}

// ---------------------------------------------------------------------------
// proj_qkv: blockIdx.x = 16-row tile (N/16 = 6250 exact), blockIdx.y = matrix
//           (0:Wq->q, 1:Wk->k, 2:Wv->v), wave-in-block = 16-col tile (8 waves).
// ---------------------------------------------------------------------------
__global__ void rnl_proj_qkv_kernel(const float* __restrict__ h,
                                    const float* __restrict__ Wq,
                                    const float* __restrict__ Wk,
                                    const float* __restrict__ Wv,
                                    float* __restrict__ q,
                                    float* __restrict__ k,
                                    float* __restrict__ v,
                                    int N) {
    const int lane = threadIdx.x & 31;
    const int wv   = threadIdx.x >> 5;        // 0..7 -> column tile
    const int row0 = blockIdx.x * 16;
    const int col0 = wv * 16;
    if (row0 >= N) return;

    const int mat = blockIdx.y;
    const float* W = (mat == 0) ? Wq : ((mat == 1) ? Wk : Wv);
    float* out     = (mat == 0) ? q  : ((mat == 1) ? k  : v);

    v8f c = wmma_tile_xwT(h, W, row0, col0, lane);

    const int half = lane >> 4;
    const int l16  = lane & 15;
    float* obase = out + (size_t)(row0 + half * 8) * DIM + col0 + l16;
#pragma unroll
    for (int r = 0; r < 8; ++r) obase[(size_t)r * DIM] = c[r];
}

// ---------------------------------------------------------------------------
// proj_out: out = relu(agg @ Wp^T + bp + h), same tiling, fused epilogue.
// ---------------------------------------------------------------------------
__global__ void rnl_proj_out_kernel(const float* __restrict__ agg,
                                    const float* __restrict__ Wp,
                                    const float* __restrict__ bp,
                                    const float* __restrict__ h,
                                    float* __restrict__ out,
                                    int N) {
    const int lane = threadIdx.x & 31;
    const int wv   = threadIdx.x >> 5;
    const int row0 = blockIdx.x * 16;
    const int col0 = wv * 16;
    if (row0 >= N) return;

    v8f c = wmma_tile_xwT(agg, Wp, row0, col0, lane);

    const int half = lane >> 4;
    const int l16  = lane & 15;
    const int col  = col0 + l16;
    const float bias = bp[col];
#pragma unroll
    for (int r = 0; r < 8; ++r) {
        const size_t row = (size_t)(row0 + half * 8 + r);
        float val = c[r] + bias + h[row * DIM + col];
        out[row * DIM + col] = fmaxf(val, 0.0f);
    }
}

// ---------------------------------------------------------------------------
// score: one wave per edge; lane l covers feature chunk [4l, 4l+4)
// ---------------------------------------------------------------------------
__global__ void rnl_score_kernel(const float* __restrict__ q,
                                 const float* __restrict__ k,
                                 const int* __restrict__ src,
                                 const int* __restrict__ dst,
                                 float* __restrict__ scores,
                                 int E) {
    const int e = blockIdx.x * (blockDim.x >> 5) + (threadIdx.x >> 5);
    if (e >= E) return;
    const int lane = threadIdx.x & 31;
    const int s = src[e];
    const int d = dst[e];

    const float4 qv = *(const float4*)(q + (size_t)d * DIM + lane * 4);
    const float4 kv = *(const float4*)(k + (size_t)s * DIM + lane * 4);
    float p = qv.x * kv.x + qv.y * kv.y + qv.z * kv.z + qv.w * kv.w;
#pragma unroll
    for (int off = 16; off > 0; off >>= 1) p += __shfl_xor(p, off, 32);

    if (lane == 0) scores[e] = p * 0.08838834764831845f;  // 1/sqrt(128)
}

// ---------------------------------------------------------------------------
// segment max via monotone float -> ordered signed-int mapping + atomicMax(i32)
// ---------------------------------------------------------------------------
__device__ __forceinline__ int f32_to_ordered(float f) {
    int i = __float_as_int(f);
    return (i >= 0) ? i : (i ^ 0x7FFFFFFF);
}
__device__ __forceinline__ float ordered_to_f32(int i) {
    return __int_as_float((i >= 0) ? i : (i ^ 0x7FFFFFFF));
}

__global__ void rnl_segmax_kernel(const float* __restrict__ scores,
                                  const int* __restrict__ dst,
                                  int* __restrict__ maxkey,
                                  int E) {
    int e = blockIdx.x * blockDim.x + threadIdx.x;
    if (e >= E) return;
    atomicMax(&maxkey[dst[e]], f32_to_ordered(scores[e]));  // global_atomic_max_i32
}

// ---------------------------------------------------------------------------
// expsum: exp(clip(score - max, -50, 50)); atomicAdd per-dst sum
// ---------------------------------------------------------------------------
__global__ void rnl_expsum_kernel(const float* __restrict__ scores,
                                  const int* __restrict__ dst,
                                  const int* __restrict__ maxkey,
                                  float* __restrict__ expb,
                                  float* __restrict__ sumv,
                                  int E) {
    int e = blockIdx.x * blockDim.x + threadIdx.x;
    if (e >= E) return;
    const int d = dst[e];
    float s = scores[e] - ordered_to_f32(maxkey[d]);
    s = fminf(fmaxf(s, -50.0f), 50.0f);
    float ex = __expf(s);
    expb[e] = ex;
    atomicAdd(&sumv[d], ex);  // global_atomic_add_f32
}

// ---------------------------------------------------------------------------
// agg: one wave per edge, agg[dst] += alpha * v[src]  (f32 atomics into L2)
// ---------------------------------------------------------------------------
__global__ void rnl_agg_kernel(const float* __restrict__ v,
                               const float* __restrict__ expb,
                               const float* __restrict__ sumv,
                               const int* __restrict__ src,
                               const int* __restrict__ dst,
                               float* __restrict__ agg,
                               int E) {
    const int e = blockIdx.x * (blockDim.x >> 5) + (threadIdx.x >> 5);
    if (e >= E) return;
    const int lane = threadIdx.x & 31;
    const int s = src[e];
    const int d = dst[e];

    float alpha = expb[e] / (sumv[d] + 1e-9f);
    if (!isfinite(alpha)) alpha = 0.0f;  // nan_to_num

    const float4 vv = *(const float4*)(v + (size_t)s * DIM + lane * 4);
    float* ab = agg + (size_t)d * DIM + lane * 4;
    atomicAdd(ab + 0, vv.x * alpha);
    atomicAdd(ab + 1, vv.y * alpha);
    atomicAdd(ab + 2, vv.z * alpha);
    atomicAdd(ab + 3, vv.w * alpha);
}

// ---------------------------------------------------------------------------
// launch
// ---------------------------------------------------------------------------
extern "C" void kernel_launch(void* const* d_in, const int* in_sizes, int n_in,
                              void* d_out, int out_size, void* d_ws, size_t ws_size,
                              hipStream_t stream) {
    const float* h     = (const float*)d_in[0];
    const int*   edges = (const int*)  d_in[1];
    const float* Wq    = (const float*)d_in[2];
    const float* Wk    = (const float*)d_in[3];
    const float* Wv    = (const float*)d_in[4];
    const float* Wp    = (const float*)d_in[5];
    const float* bp    = (const float*)d_in[6];

    const int N = in_sizes[0] / DIM;   // 100000 (multiple of 16)
    const int E = in_sizes[1] / 2;     // 600000
    const int* src = edges;            // edges[0, :]
    const int* dst = edges + E;        // edges[1, :]

    const size_t Nf = (size_t)N * DIM;

    // workspace layout (floats): q | k | v | agg | scores | expb | sumv | maxkey
    float* q      = (float*)d_ws;
    float* k      = q + Nf;
    float* v      = k + Nf;
    float* agg    = v + Nf;
    float* scores = agg + Nf;
    float* expb   = scores + E;
    float* sumv   = expb + E;
    int*   maxkey = (int*)(sumv + N);

    // 1) init accumulators (workspace is poisoned)
    {
        int threads = 256;
        int blocks  = (int)((Nf + threads - 1) / threads);
        rnl_init_kernel<<<blocks, threads, 0, stream>>>(agg, maxkey, sumv, N);
    }
    // 2) q,k,v projections (WMMA). grid: (row tiles, 3 matrices); 8 waves = 8 col tiles
    {
        dim3 grid((N + 15) / 16, 3);
        rnl_proj_qkv_kernel<<<grid, 256, 0, stream>>>(h, Wq, Wk, Wv, q, k, v, N);
    }
    // 3) per-edge scores (1 wave per edge, 8 edges per block)
    {
        int blocks = (E + 7) / 8;
        rnl_score_kernel<<<blocks, 256, 0, stream>>>(q, k, src, dst, scores, E);
    }
    // 4) segment max
    {
        int blocks = (E + 255) / 256;
        rnl_segmax_kernel<<<blocks, 256, 0, stream>>>(scores, dst, maxkey, E);
    }
    // 5) exp + segment sum
    {
        int blocks = (E + 255) / 256;
        rnl_expsum_kernel<<<blocks, 256, 0, stream>>>(scores, dst, maxkey, expb, sumv, E);
    }
    // 6) weighted aggregation
    {
        int blocks = (E + 7) / 8;
        rnl_agg_kernel<<<blocks, 256, 0, stream>>>(v, expb, sumv, src, dst, agg, E);
    }
    // 7) output projection + bias + residual + relu (WMMA, fused epilogue)
    {
        dim3 grid((N + 15) / 16);
        rnl_proj_out_kernel<<<grid, 256, 0, stream>>>(agg, Wp, bp, h, (float*)d_out, N);
    }
}